// RADaugment_71382356459757
// MI455X (gfx1250) — compile-verified
//
#include <hip/hip_runtime.h>
#include <stdint.h>

// Problem geometry (from reference)
#define H_    155
#define W_    86
#define CH_   147          // int(155*0.95)
#define CW_   81           // int(86*0.95)
#define NPIX  (H_ * W_)    // 13330
#define NCROP (CH_ * CW_)  // 11907
#define C_    4            // channels (fixed by reference)
#define NTHREADS 256

typedef uint32_t u32x4 __attribute__((ext_vector_type(4)));
typedef uint32_t u32x8 __attribute__((ext_vector_type(8)));

__device__ __forceinline__ float clamp01(float v) {
    return fminf(fmaxf(v, 0.0f), 1.0f);
}

// Bilinear sample of the LDS crop tile at output pixel p, post-brightness+clip.
__device__ __forceinline__ float sample_px(const float* __restrict__ sCrop, int p,
                                           float brightness) {
    const float KY = (float)(147.0 / 155.0);
    const float KX = (float)(81.0 / 86.0);
    int oy = p / W_;                 // magic-multiply, no hw divide
    int ox = p - oy * W_;
    float sy = fmaxf((oy + 0.5f) * KY - 0.5f, 0.0f);
    float sx = fmaxf((ox + 0.5f) * KX - 0.5f, 0.0f);
    int y0 = (int)sy;
    int x0 = (int)sx;
    float wy = sy - (float)y0;
    float wx = sx - (float)x0;
    int y1 = (y0 + 1 < CH_) ? y0 + 1 : CH_ - 1;
    int x1 = (x0 + 1 < CW_) ? x0 + 1 : CW_ - 1;
    float v00 = sCrop[y0 * CW_ + x0];
    float v01 = sCrop[y0 * CW_ + x1];
    float v10 = sCrop[y1 * CW_ + x0];
    float v11 = sCrop[y1 * CW_ + x1];
    float r0 = v00 + (v10 - v00) * wy;  // interp along y at x0
    float r1 = v01 + (v11 - v01) * wy;  // interp along y at x1
    float v  = r0 + (r1 - r0) * wx;     // interp along x
    return clamp01(v * brightness);
}

__global__ __launch_bounds__(NTHREADS)
void RADaugment_fused_kernel(const float* __restrict__ x,
                             const int* __restrict__ top,
                             const int* __restrict__ left,
                             const float* __restrict__ bright_u,
                             const float* __restrict__ contrast_u,
                             const float* __restrict__ noise,
                             float* __restrict__ out) {
    extern __shared__ float smem[];
    float* sCrop  = smem;                 // NCROP floats: TDM destination
    float* sNoise = smem + NCROP;         // NPIX  floats: async-load destination
    float* sRed   = smem + NCROP + NPIX;  // NTHREADS floats: reduction

    const int bc  = blockIdx.x;   // one (b, c) image per workgroup
    const int b   = bc >> 2;      // C_ == 4
    const int tid = threadIdx.x;

    const int t = __builtin_amdgcn_readfirstlane(top[b]);
    const int l = __builtin_amdgcn_readfirstlane(left[b]);

    // ---- Phase 0a: TDM DMA of the 147x81 crop tile (stride 86) into LDS ----
    // Issued by wave 0 only (TDM ignores EXEC; duplicate issue per wave would
    // duplicate the DMA). Descriptor per CDNA5 ISA ch.8 (D# groups 0 & 1).
    if (tid < 32) {
        const float* gsrc = x + ((size_t)bc * H_ + (size_t)t) * W_ + (size_t)l;
        uint64_t ga   = (uint64_t)(uintptr_t)gsrc;
        uint32_t galo = __builtin_amdgcn_readfirstlane((uint32_t)ga);
        uint32_t gahi = __builtin_amdgcn_readfirstlane((uint32_t)(ga >> 32));
        uint32_t lofs = __builtin_amdgcn_readfirstlane((uint32_t)(uintptr_t)sCrop);

        u32x4 g0;
        g0[0] = 1u;                                   // count=1, is_restore=0, gather off
        g0[1] = lofs;                                 // lds_addr (bytes)
        g0[2] = galo;                                 // global_addr[31:0]
        g0[3] = (gahi & 0x01FFFFFFu) | (2u << 30);    // global_addr[56:32] | type=2

        u32x8 g1;
        g1[0] = (2u << 16);                           // wg_mask=0, data_size=4B
        g1[1] = ((uint32_t)W_ & 0xFFFFu) << 16;       // tensor_dim0 = 86
        g1[2] = (((uint32_t)H_) & 0xFFFFu) << 16;     // tensor_dim0 hi=0 | tensor_dim1=155
        g1[3] = ((uint32_t)CW_) << 16;                // tensor_dim1 hi=0 | tile_dim0=81
        g1[4] = (uint32_t)CH_;                        // tile_dim1=147, tile_dim2=0
        g1[5] = (uint32_t)W_;                         // tensor_dim0_stride = 86
        g1[6] = ((uint32_t)NPIX & 0xFFFFu) << 16;     // stride0 hi=0 | tensor_dim1_stride lo
        g1[7] = ((uint32_t)NPIX) >> 16;               // tensor_dim1_stride hi

        asm volatile("tensor_load_to_lds %0, %1" :: "s"(g0), "s"(g1) : "memory");
    }

    // ---- Phase 0b: async per-lane prefetch of this image's noise into LDS ----
    // Each thread prefetches exactly the elements it will read in phase 3, so a
    // per-wave s_wait_asynccnt suffices (no cross-wave dependency). B32 because
    // the per-image base (bc*13330 elements) is only 8B-aligned for odd bc.
    {
        uint32_t nofs = (uint32_t)(uintptr_t)sNoise;
        const float* nsrc = noise + (size_t)bc * NPIX;
        for (int p = tid; p < NPIX; p += NTHREADS) {
            uint32_t la = nofs + (uint32_t)p * 4u;
            uint64_t ga = (uint64_t)(uintptr_t)(nsrc + p);
            asm volatile("global_load_async_to_lds_b32 %0, %1, off"
                         :: "v"(la), "v"(ga) : "memory");
        }
    }

    if (tid < 32) {
        __builtin_amdgcn_s_wait_tensorcnt(0);  // crop tile landed in LDS
    }
    __syncthreads();                           // publish to all waves

    const float brightness = 0.7f + bright_u[0] * 0.6f;
    const float contrast   = 0.8f + contrast_u[0] * 0.4f;

    // ---- Phase 1: resize + brightness, accumulate sum for the mean ----
    float sum = 0.0f;
    for (int p = tid; p < NPIX; p += NTHREADS) {
        sum += sample_px(sCrop, p, brightness);
    }

    // ---- Phase 2: deterministic LDS tree reduction -> mean ----
    sRed[tid] = sum;
    __syncthreads();
    #pragma unroll
    for (int s = NTHREADS / 2; s > 0; s >>= 1) {
        if (tid < s) sRed[tid] += sRed[tid + s];
        __syncthreads();
    }
    const float mean = sRed[0] * (1.0f / (float)NPIX);

    // ---- Phase 3: contrast + noise + clip, write out (non-temporal) ----
    asm volatile("s_wait_asynccnt 0" ::: "memory");  // my wave's noise is in LDS

    float* oBase = out + (size_t)bc * NPIX;
    for (int p = tid; p < NPIX; p += NTHREADS) {
        float val = sample_px(sCrop, p, brightness);          // recompute (LDS-cheap)
        val = clamp01((val - mean) * contrast + mean);
        val = clamp01(val + sNoise[p] * 0.02f);
        __builtin_nontemporal_store(val, oBase + p);          // never re-read
    }
}

extern "C" void kernel_launch(void* const* d_in, const int* in_sizes, int n_in,
                              void* d_out, int out_size, void* d_ws, size_t ws_size,
                              hipStream_t stream) {
    const float* x     = (const float*)d_in[0];
    const int*   top   = (const int*)d_in[1];
    const int*   left  = (const int*)d_in[2];
    const float* bu    = (const float*)d_in[3];
    const float* cu    = (const float*)d_in[4];
    const float* noise = (const float*)d_in[5];
    float*       out   = (float*)d_out;

    const int BC = in_sizes[0] / NPIX;        // B*C = 4096

    const size_t smem_bytes = (size_t)(NCROP + NPIX + NTHREADS) * sizeof(float);
    RADaugment_fused_kernel<<<BC, NTHREADS, smem_bytes, stream>>>(
        x, top, left, bu, cu, noise, out);
}